// stagenet_50895362458122
// MI455X (gfx1250) — compile-verified
//
#include <hip/hip_runtime.h>
#include <math.h>

// ---------------- problem constants ----------------
#define HH 192
#define WW 192
#define HWN (192*192)
#define CN 32
#define GN 8
#define DN 32
#define NSRC 4

// output tuple offsets (flat concat in return order)
#define OFF_DEPTH  0
#define OFF_CONF   (HWN)
#define OFF_ATTN   (2*HWN)
#define OFF_ENORM  (OFF_ATTN + DN*HWN)
#define OFF_SUMW   (OFF_ENORM + 3*HWN)
#define OFF_WEIGHT (OFF_SUMW + HWN)
#define OFF_NVALID (OFF_WEIGHT + NSRC*HWN)
#define OFF_MIND   (OFF_NVALID + DN*HWN)
#define OFF_MAXD   (OFF_MIND + HWN)

// 4-byte-aligned float pair for paired x-tap gathers (B64 when HW allows)
struct __attribute__((packed, aligned(4))) F2 { float x, y; };

// ---------------- helpers ----------------
__device__ __forceinline__ unsigned fkey(float f) {
    unsigned u = __float_as_uint(f);
    return (u & 0x80000000u) ? ~u : (u | 0x80000000u);
}
__device__ __forceinline__ float funkey(unsigned u) {
    return (u & 0x80000000u) ? __uint_as_float(u & 0x7fffffffu)
                             : __uint_as_float(~u);
}
__device__ __forceinline__ float wsum32(float v) {
    #pragma unroll
    for (int o = 16; o > 0; o >>= 1) v += __shfl_xor(v, o, 32);
    return v;
}
__device__ __forceinline__ float wmax32(float v) {
    #pragma unroll
    for (int o = 16; o > 0; o >>= 1) v = fmaxf(v, __shfl_xor(v, o, 32));
    return v;
}

__device__ void inv4x4(const float a[16], float out[16]) {
    float aug[4][8];
    for (int i = 0; i < 4; i++)
        for (int j = 0; j < 4; j++) { aug[i][j] = a[i*4+j]; aug[i][4+j] = (i==j) ? 1.f : 0.f; }
    for (int col = 0; col < 4; col++) {
        int piv = col; float best = fabsf(aug[col][col]);
        for (int r = col+1; r < 4; r++) { float v = fabsf(aug[r][col]); if (v > best) { best = v; piv = r; } }
        if (piv != col)
            for (int j = 0; j < 8; j++) { float t = aug[col][j]; aug[col][j] = aug[piv][j]; aug[piv][j] = t; }
        float idv = 1.f / aug[col][col];
        for (int j = 0; j < 8; j++) aug[col][j] *= idv;
        for (int r = 0; r < 4; r++) {
            if (r == col) continue;
            float f = aug[r][col];
            for (int j = 0; j < 8; j++) aug[r][j] -= f * aug[col][j];
        }
    }
    for (int i = 0; i < 4; i++)
        for (int j = 0; j < 4; j++) out[i*4+j] = aug[i][4+j];
}

// ---------------- kernel 1: projection / normal-vector setup (1 thread) ----------------
// ws float layout: view i at [i*16]: rot[9], trans[3], norv[3], pad
// ws uint [64]=mapped global min, [65]=mapped global max
__global__ void setup_kernel(const float* __restrict__ proj,
                             const float* __restrict__ rot,
                             float* __restrict__ wsf,
                             unsigned* __restrict__ wsu) {
    if (threadIdx.x != 0 || blockIdx.x != 0) return;
    // ref_proj_new = [K0[:3,:3] @ E0[:3,:4] ; E0 row3]
    const float* E0 = proj;        // (1,V,2,4,4): view0, mat0
    const float* K0 = proj + 16;   // view0, mat1
    float refPN[16];
    for (int r = 0; r < 3; r++)
        for (int c = 0; c < 4; c++) {
            float s = 0.f;
            for (int k = 0; k < 3; k++) s += K0[r*4+k] * E0[k*4+c];
            refPN[r*4+c] = s;
        }
    for (int c = 0; c < 4; c++) refPN[12+c] = E0[12+c];
    float Minv[16];
    inv4x4(refPN, Minv);
    // inv of ref rotation (3x3 adjugate)
    const float* R0 = rot;
    float det = R0[0]*(R0[4]*R0[8]-R0[5]*R0[7])
              - R0[1]*(R0[3]*R0[8]-R0[5]*R0[6])
              + R0[2]*(R0[3]*R0[7]-R0[4]*R0[6]);
    float id = 1.f / det;
    float Ri[9];
    Ri[0]=(R0[4]*R0[8]-R0[5]*R0[7])*id;  Ri[1]=(R0[2]*R0[7]-R0[1]*R0[8])*id;  Ri[2]=(R0[1]*R0[5]-R0[2]*R0[4])*id;
    Ri[3]=(R0[5]*R0[6]-R0[3]*R0[8])*id;  Ri[4]=(R0[0]*R0[8]-R0[2]*R0[6])*id;  Ri[5]=(R0[2]*R0[3]-R0[0]*R0[5])*id;
    Ri[6]=(R0[3]*R0[7]-R0[4]*R0[6])*id;  Ri[7]=(R0[1]*R0[6]-R0[0]*R0[7])*id;  Ri[8]=(R0[0]*R0[4]-R0[1]*R0[3])*id;

    for (int i = 0; i < NSRC; i++) {
        int v = i + 1;
        const float* Es = proj + v*32;
        const float* Ks = proj + v*32 + 16;
        float sPN[16];
        for (int r = 0; r < 3; r++)
            for (int c = 0; c < 4; c++) {
                float s = 0.f;
                for (int k = 0; k < 3; k++) s += Ks[r*4+k] * Es[k*4+c];
                sPN[r*4+c] = s;
            }
        for (int c = 0; c < 4; c++) sPN[12+c] = Es[12+c];
        // P = sPN @ Minv, rows 0..2; rot = P[:3,:3], trans = P[:3,3]
        for (int r = 0; r < 3; r++) {
            float pr[4];
            for (int c = 0; c < 4; c++) {
                float s = 0.f;
                for (int k = 0; k < 4; k++) s += sPN[r*4+k] * Minv[k*4+c];
                pr[c] = s;
            }
            wsf[i*16 + r*3+0] = pr[0];
            wsf[i*16 + r*3+1] = pr[1];
            wsf[i*16 + r*3+2] = pr[2];
            wsf[i*16 + 9 + r] = pr[3];
        }
        // norv = row 2 of (R_v @ R0^-1)
        const float* Rv = rot + v*9;
        for (int c = 0; c < 3; c++)
            wsf[i*16 + 12 + c] = Rv[6]*Ri[c] + Rv[7]*Ri[3+c] + Rv[8]*Ri[6+c];
        wsf[i*16 + 15] = 0.f;
    }
    wsu[64] = 0xFFFFFFFFu;  // mapped-min identity
    wsu[65] = 0u;           // mapped-max identity
}

// ---------------- kernel 2: global min/max of depth_hypo ----------------
__global__ __launch_bounds__(256) void minmax_kernel(const float* __restrict__ dh,
                                                     unsigned* __restrict__ wsu) {
    const int n = DN * HWN;
    float lmin = INFINITY, lmax = -INFINITY;
    for (int i = blockIdx.x * blockDim.x + threadIdx.x; i < n; i += gridDim.x * blockDim.x) {
        float v = dh[i];
        lmin = fminf(lmin, v);
        lmax = fmaxf(lmax, v);
    }
    #pragma unroll
    for (int o = 16; o > 0; o >>= 1) {
        lmin = fminf(lmin, __shfl_xor(lmin, o, 32));
        lmax = fmaxf(lmax, __shfl_xor(lmax, o, 32));
    }
    if ((threadIdx.x & 31) == 0) {
        atomicMin(&wsu[64], fkey(lmin));
        atomicMax(&wsu[65], fkey(lmax));
    }
}

// ---------------- kernel 3: main per-pixel pipeline (1 wave = 1 pixel, lane = depth) ----------------
__global__ __launch_bounds__(256) void main_kernel(
    const float* __restrict__ ref, const float* __restrict__ src,
    const float* __restrict__ normal, const float* __restrict__ dh,
    const float* __restrict__ wreg, const float* __restrict__ wsf,
    const unsigned* __restrict__ wsu, float* __restrict__ out) {

    const int lane = threadIdx.x & 31;             // lane == depth index (D == 32)
    const int p = blockIdx.x * 8 + (threadIdx.x >> 5);
    const int h = p / WW, w = p - h * WW;
    const float xf = (float)w, yf = (float)h;

    const float myDepth = dh[lane * HWN + p];
    const float refc = ref[lane * HWN + p];        // lane holds ref channel == lane
    const float npl0 = normal[p], npl1 = normal[HWN + p], npl2 = normal[2*HWN + p];

    const float hmin = funkey(wsu[64]);
    const float hmax = funkey(wsu[65]);

    float cf[GN];
    #pragma unroll
    for (int g = 0; g < GN; g++) cf[g] = 0.f;
    float cws = 1e-8f, swsum = 1e-8f, nvalid = 0.f;

    for (int i = 0; i < NSRC; i++) {
        const float* M = wsf + i * 16;
        const float r00=M[0], r01=M[1], r02=M[2], r10=M[3], r11=M[4], r12=M[5],
                    r20=M[6], r21=M[7], r22=M[8], t0=M[9], t1=M[10], t2=M[11];
        const float srcw = fmaxf(npl0*M[12] + npl1*M[13] + npl2*M[14], 0.f) + 0.01f;

        // homography: (rot @ [x,y,1]) * depth + trans
        const float rx = r00*xf + r01*yf + r02;
        const float ry = r10*xf + r11*yf + r12;
        const float rz = r20*xf + r21*yf + r22;
        float pz = fmaf(rz, myDepth, t2);
        if (pz == 0.f) pz = 1e-9f;
        const float invz = 1.f / pz;
        const float px = fmaf(rx, myDepth, t0) * invz;
        const float py = fmaf(ry, myDepth, t1) * invz;

        const float x0f = floorf(px), y0f = floorf(py);
        const float wx = px - x0f, wy = py - y0f;
        const float x1f = x0f + 1.f, y1f = y0f + 1.f;
        const float ibx0 = (x0f >= 0.f && x0f <= (float)(WW-1)) ? 1.f : 0.f;
        const float ibx1 = (x1f >= 0.f && x1f <= (float)(WW-1)) ? 1.f : 0.f;
        const float iby0 = (y0f >= 0.f && y0f <= (float)(HH-1)) ? 1.f : 0.f;
        const float iby1 = (y1f >= 0.f && y1f <= (float)(HH-1)) ? 1.f : 0.f;
        const float w00 = (1.f-wx)*(1.f-wy) * (ibx0*iby0);
        const float w10 = wx*(1.f-wy)       * (ibx1*iby0);
        const float w01 = (1.f-wx)*wy       * (ibx0*iby1);
        const float w11 = wx*wy             * (ibx1*iby1);
        const int ix0 = (int)fminf(fmaxf(x0f, 0.f), (float)(WW-1));
        const int ix1 = (int)fminf(fmaxf(x1f, 0.f), (float)(WW-1));
        const int iy0 = (int)fminf(fmaxf(y0f, 0.f), (float)(HH-1));
        const int iy1 = (int)fminf(fmaxf(y1f, 0.f), (float)(HH-1));

        // Pair the two x-taps of each row into one 8-byte gather.
        // Collapsed-clip case (ix1==ix0): both taps read the same texel -> fold
        // weights onto the low element. If the pair base is the plane's last
        // texel, shift base down one and move the folded weight to the high
        // element (exact, since the hi-tap weight is zero in that case).
        const bool eqx = (ix1 == ix0);
        const float wlo0 = w00 + (eqx ? w10 : 0.f);
        const float whi0 = eqx ? 0.f : w10;
        const float wlo1 = w01 + (eqx ? w11 : 0.f);
        const float whi1 = eqx ? 0.f : w11;
        int b0 = iy0*WW + ix0;
        int b1 = iy1*WW + ix0;
        const int s0 = (b0 == HWN-1) ? 1 : 0;
        const int s1 = (b1 == HWN-1) ? 1 : 0;
        b0 -= s0; b1 -= s1;
        const float wa0 = s0 ? 0.f  : wlo0;
        const float wa1 = s0 ? wlo0 : whi0;
        const float wb0 = s1 ? 0.f  : wlo1;
        const float wb1 = s1 ? wlo1 : whi1;

        const float* sp = src + (size_t)i * CN * HWN;
        float corf[GN];
        #pragma unroll
        for (int g = 0; g < GN; g++) corf[g] = 0.f;
        float v0 = 0.f;
        #pragma unroll 8
        for (int c = 0; c < CN; c++) {
            const float* spc = sp + c * HWN;
            const F2 A  = *(const F2*)(spc + b0);
            const F2 Bv = *(const F2*)(spc + b1);
            float v = fmaf(A.x, wa0,
                      fmaf(A.y, wa1,
                      fmaf(Bv.x, wb0, Bv.y * wb1)));
            if (c == 0) v0 = v;
            const float rc = __shfl(refc, c, 32);  // broadcast ref channel c
            corf[c >> 2] = fmaf(rc, v, corf[c >> 2]);
        }
        const float valid = (v0 != 0.f) ? 1.f : 0.f;
        nvalid += valid;
        const float swd = srcw * valid;
        swsum += swd;

        const float vsum = wsum32(valid);  // sims[i] = srcw * mean_d(valid)
        if (lane == 0) out[OFF_WEIGHT + i*HWN + p] = srcw * (vsum * (1.f / (float)DN));

        float corsum = 0.f;
        #pragma unroll
        for (int g = 0; g < GN; g++) { corf[g] *= 0.25f; corsum += corf[g]; }

        // cor_w = softmax_d(corsum) / sqrt(C)
        const float m = wmax32(corsum);
        const float e = expf(corsum - m);
        const float s = wsum32(e);
        const float cw = (e / s) * 0.17677669529663688f;  // 1/sqrt(32)
        cws += cw;
        const float f = cw * swd;
        #pragma unroll
        for (int g = 0; g < GN; g++) cf[g] = fmaf(f, corf[g], cf[g]);
    }

    // attn_logits = cor_feats . w_reg + norm_hypo
    const float inv = 1.f / (swsum * cws);
    float logit = (myDepth - hmin) / (hmax - hmin);
    #pragma unroll
    for (int g = 0; g < GN; g++) logit = fmaf(cf[g] * inv, wreg[g], logit);

    const float lm = wmax32(logit);
    const float le = expf(logit - lm);
    const float ls = wsum32(le);
    const float aw = le / ls;
    out[OFF_ATTN + lane*HWN + p] = aw;
    out[OFF_NVALID + lane*HWN + p] = nvalid;

    // argmax over depths (first index on ties, like jnp.argmax)
    float bv = aw; int bi = lane;
    #pragma unroll
    for (int o = 16; o > 0; o >>= 1) {
        const float ov = __shfl_xor(bv, o, 32);
        const int   oi = __shfl_xor(bi, o, 32);
        if (ov > bv || (ov == bv && oi < bi)) { bv = ov; bi = oi; }
    }
    const float dsel = __shfl(myDepth, bi, 32);
    const float d0 = __shfl(myDepth, 0, 32);
    const float d1 = __shfl(myDepth, 1, 32);
    if (lane == 0) {
        const float itv = d1 - d0;
        out[OFF_DEPTH + p] = dsel;
        out[OFF_CONF  + p] = bv;
        out[OFF_MIND  + p] = dsel - itv;
        out[OFF_MAXD  + p] = dsel + itv;
        out[OFF_SUMW  + p] = swsum * (1.f / (float)NSRC);  // sum_weight[:,0,0] at d==0
    }
}

// ---------------- kernel 4: est_normal_plane = w_norm(3x3) @ normal(3xHW) via WMMA f32 16x16x4 ----------------
typedef float v2f __attribute__((ext_vector_type(2)));
typedef float v8f __attribute__((ext_vector_type(8)));

__global__ __launch_bounds__(256) void enorm_kernel(const float* __restrict__ normal,
                                                    const float* __restrict__ wnorm,
                                                    float* __restrict__ out) {
    const int lane = threadIdx.x & 31;
    const int wave = threadIdx.x >> 5;
    const int base = (blockIdx.x * 8 + wave) * 16;   // 16 pixels per wave
    const int n = lane & 15;
    const int kp = lane >> 4;                         // lanes 0-15: K={0,1}; 16-31: K={2,3}
    const int k0 = 2 * kp;
    const int pix = base + n;

    // A (16x4): rows 0..2 = w_norm, rest zero (lane L holds M=L%16, K per kp)
    v2f a;
    a.x = (n < 3)            ? wnorm[n*3 + k0]     : 0.f;
    a.y = (n < 3 && k0 < 2)  ? wnorm[n*3 + k0 + 1] : 0.f;
    // B (4x16): rows 0..2 = normal_plane channels for 16 pixels, row 3 zero.
    // Load unconditionally from a valid channel, then select (avoid exec branch).
    const float t = normal[(kp ? 2 : 1) * HWN + pix];
    v2f b;
    b.x = normal[k0 * HWN + pix];                    // k0 = 0 or 2
    b.y = kp ? 0.f : t;                              // k = 1 live, k = 3 zero

    v8f c = {0.f, 0.f, 0.f, 0.f, 0.f, 0.f, 0.f, 0.f};
    v8f dmat = __builtin_amdgcn_wmma_f32_16x16x4_f32(
        /*neg_a=*/false, a, /*neg_b=*/false, b,
        /*c_mod=*/(short)0, c, /*reuse_a=*/false, /*reuse_b=*/false);

    if (lane < 16) {   // lanes 0-15 hold D rows 0..7; rows 0..2 are the outputs
        out[OFF_ENORM + 0*HWN + pix] = dmat[0];
        out[OFF_ENORM + 1*HWN + pix] = dmat[1];
        out[OFF_ENORM + 2*HWN + pix] = dmat[2];
    }
}

// ---------------- launcher ----------------
extern "C" void kernel_launch(void* const* d_in, const int* in_sizes, int n_in,
                              void* d_out, int out_size, void* d_ws, size_t ws_size,
                              hipStream_t stream) {
    const float* ref      = (const float*)d_in[0];  // (1,32,192,192)
    const float* src      = (const float*)d_in[1];  // (4,1,32,192,192)
    const float* rotation = (const float*)d_in[2];  // (1,5,3,3)
    const float* normal   = (const float*)d_in[3];  // (1,3,192,192)
    const float* proj     = (const float*)d_in[4];  // (1,5,2,4,4)
    const float* dh       = (const float*)d_in[5];  // (1,32,192,192)
    const float* wreg     = (const float*)d_in[6];  // (8,)
    const float* wnorm    = (const float*)d_in[7];  // (3,3)
    float* out = (float*)d_out;
    float* wsf = (float*)d_ws;
    unsigned* wsu = (unsigned*)d_ws;

    setup_kernel<<<1, 1, 0, stream>>>(proj, rotation, wsf, wsu);
    minmax_kernel<<<256, 256, 0, stream>>>(dh, wsu);
    main_kernel<<<HWN / 8, 256, 0, stream>>>(ref, src, normal, dh, wreg, wsf, wsu, out);
    enorm_kernel<<<HWN / 128, 256, 0, stream>>>(normal, wnorm, out);
}